// TimingNetWirelength_90091234001231
// MI455X (gfx1250) — compile-verified
//
#include <hip/hip_runtime.h>
#include <stdint.h>

// ---------------------------------------------------------------------------
// TimingNetWirelength on MI455X (gfx1250)
// Memory-bound gather+reduce: ~88 MB HBM traffic -> ~3.8us floor @ 23.3 TB/s.
// CDNA5 path: TDM (tensor_load_to_lds, TENSORcnt) stages the streaming
// index/weight tiles into LDS, freeing the vector-memory pipeline for the
// random pos[] gathers (L2-resident: 40 MB << 192 MB L2). Gather loop is
// unrolled x4 for memory-level parallelism.
// ---------------------------------------------------------------------------

#define TILE      2048          // tnets per block
#define BLOCK     256           // 8 wave32 waves

typedef __attribute__((ext_vector_type(4))) unsigned int u32x4;
typedef __attribute__((ext_vector_type(8))) unsigned int u32x8;

// Issue a 1-D TDM copy of n_dwords DWORDs from global -> LDS.
// Descriptor layout per CDNA5 ISA (#8.3 group0, #8.4 group1).
__device__ __forceinline__ void tdm_load_1d(unsigned lds_byte_addr,
                                            const void* gptr,
                                            unsigned n_dwords) {
  unsigned long long ga = (unsigned long long)(uintptr_t)gptr;

  u32x4 g0;
  g0[0] = 0x1u;                                   // count=1, user mode, no gather
  g0[1] = lds_byte_addr;                          // lds_addr (bytes)
  g0[2] = (unsigned)(ga & 0xFFFFFFFFu);           // global_addr[31:0]
  g0[3] = (unsigned)((ga >> 32) & 0x01FFFFFFu)    // global_addr[56:32]
        | 0x80000000u;                            // type = 2 ("image")

  u32x8 g1;
  g1[0] = 0x00020000u;                            // wg_mask=0, data_size=2 (4B)
  g1[1] = (n_dwords & 0xFFFFu) << 16;             // tensor_dim0[15:0]   @ bits 63:48
  g1[2] = ((n_dwords >> 16) & 0xFFFFu)            // tensor_dim0[31:16]  @ bits 79:64
        | (1u << 16);                             // tensor_dim1 = 1     @ bits 111:80
  g1[3] = (n_dwords & 0xFFFFu) << 16;             // tile_dim0           @ bits 127:112
  g1[4] = 1u;                                     // tile_dim1=1, tile_dim2=0
  g1[5] = n_dwords;                               // tensor_dim0_stride[31:0]
  g1[6] = 0u;
  g1[7] = 0u;

  u32x4 gz; gz[0] = 0u; gz[1] = 0u; gz[2] = 0u; gz[3] = 0u;  // groups 2/3 unused

  asm volatile("tensor_load_to_lds %0, %1, %2, %3"
               :: "s"(g0), "s"(g1), "s"(gz), "s"(gz)
               : "memory");
}

// Reference-faithful weighted-average span (INV_GAMMA = 1/0.5 = 2).
__device__ __forceinline__ float wavg_span(float v0, float v1) {
  const float ig = 2.0f;
  float s0 = v0 * ig, s1 = v1 * ig;
  float mx = fmaxf(s0, s1);
  float mn = fminf(s0, s1);
  float ep0 = __expf(s0 - mx), ep1 = __expf(s1 - mx);
  float en0 = __expf(mn - s0), en1 = __expf(mn - s1);
  float wa_max = (v0 * ep0 + v1 * ep1) / (ep0 + ep1);
  float wa_min = (v0 * en0 + v1 * en1) / (en0 + en1);
  return wa_max - wa_min;
}

__global__ void __launch_bounds__(BLOCK)
wl_partial_kernel(const float* __restrict__ pos,
                  const int*   __restrict__ tnetpin,   // [num_tnets*2]
                  const float* __restrict__ tw,        // [num_tnets]
                  int num_pins, int num_tnets,
                  float* __restrict__ partials) {
  __shared__ __align__(16) int   s_idx[TILE * 2];   // 16 KB
  __shared__ __align__(16) float s_w[TILE];         //  8 KB
  __shared__ float s_red[BLOCK];

  const int tile_start = blockIdx.x * TILE;
  const int cnt = min(TILE, num_tnets - tile_start);

  // Wave 0 stages this block's streaming data into LDS via the TDM.
  if (threadIdx.x < 32) {
    unsigned lds_idx = (unsigned)(uintptr_t)&s_idx[0];
    unsigned lds_w   = (unsigned)(uintptr_t)&s_w[0];
    tdm_load_1d(lds_idx, tnetpin + (size_t)tile_start * 2, (unsigned)(cnt * 2));
    tdm_load_1d(lds_w,   tw + tile_start,                  (unsigned)cnt);
    __builtin_amdgcn_s_wait_tensorcnt(0);
  }
  __syncthreads();

  const float* xs = pos;
  const float* ys = pos + num_pins;
  const int2*  sp = (const int2*)s_idx;

  float acc = 0.0f;
  // TILE/BLOCK = 8 iterations; unroll x4 to keep ~16 gathers in flight.
  #pragma unroll 4
  for (int k = threadIdx.x; k < cnt; k += BLOCK) {
    int2  p = sp[k];                 // one ds_load_b64
    float w = s_w[k];
    float x0 = xs[p.x], x1 = xs[p.y];
    float y0 = ys[p.x], y1 = ys[p.y];
    acc += w * (wavg_span(x0, x1) + wavg_span(y0, y1));
  }

  // Deterministic block reduction in LDS.
  s_red[threadIdx.x] = acc;
  __syncthreads();
  for (int s = BLOCK / 2; s > 0; s >>= 1) {
    if (threadIdx.x < s) s_red[threadIdx.x] += s_red[threadIdx.x + s];
    __syncthreads();
  }
  if (threadIdx.x == 0) partials[blockIdx.x] = s_red[0];
}

__global__ void __launch_bounds__(BLOCK)
wl_final_kernel(const float* __restrict__ partials, int n, float* __restrict__ out) {
  __shared__ float s_red[BLOCK];
  float a = 0.0f;
  for (int i = threadIdx.x; i < n; i += BLOCK) a += partials[i];
  s_red[threadIdx.x] = a;
  __syncthreads();
  for (int s = BLOCK / 2; s > 0; s >>= 1) {
    if (threadIdx.x < s) s_red[threadIdx.x] += s_red[threadIdx.x + s];
    __syncthreads();
  }
  if (threadIdx.x == 0) out[0] = s_red[0];
}

extern "C" void kernel_launch(void* const* d_in, const int* in_sizes, int n_in,
                              void* d_out, int out_size, void* d_ws, size_t ws_size,
                              hipStream_t stream) {
  const float* pos     = (const float*)d_in[0];  // [2*num_pins]
  const int*   tnetpin = (const int*)d_in[1];    // [num_tnets, 2]
  const float* tw      = (const float*)d_in[2];  // [num_tnets]
  // d_in[3] = pin_mask: unused by the reference.

  const int num_pins  = in_sizes[0] / 2;
  const int num_tnets = in_sizes[2];

  float* partials = (float*)d_ws;

  const int nblocks = (num_tnets + TILE - 1) / TILE;
  wl_partial_kernel<<<nblocks, BLOCK, 0, stream>>>(pos, tnetpin, tw,
                                                   num_pins, num_tnets, partials);
  wl_final_kernel<<<1, BLOCK, 0, stream>>>(partials, nblocks, (float*)d_out);
}